// luong_global_attention_59279138619484
// MI455X (gfx1250) — compile-verified
//
#include <hip/hip_runtime.h>
#include <math.h>

// Luong dot attention, fused single-pass (online softmax), fp32.
// B=256, L=2048, D=1024.  Memory-bound: one streaming pass over encoder_output
// (2.15 GB -> ~92us at 23.3 TB/s).  Context accumulation runs on the matrix
// pipe via V_WMMA_F32_16X16X4_F32 (natural layout, no transpose needed),
// dot products + softmax on VALU, tile buffered in LDS between the two uses.
//
// Round-2 changes: decoder_h fragments moved from 64 VGPRs into LDS, phase-C
// scheduling fenced every 4 tiles, and VGPRs capped at 256 to kill the
// s_set_vgpr_msb churn seen in round 1 and raise occupancy.

#define B_N 256
#define L_N 2048
#define D_N 1024
#define WAVES 8

typedef float v2f __attribute__((ext_vector_type(2)));
typedef float v4f __attribute__((ext_vector_type(4)));
typedef float v8f __attribute__((ext_vector_type(8)));

// ---------------------------------------------------------------------------
// Kernel 1: per (batch, part) flash-attention pass.
//   grid = B * split, block = 256 (8 waves).  Each wave processes rows in
//   groups of 4 (the WMMA K dimension).
//
// Data layout trick: global load with per-lane address
//   row = r0 + (lane<16 ? 0 : 2),  d = 64j + 4*(lane&15) .. +3   (b128)
// yields, per component c, exactly the WMMA B-fragment VGPR0 (K=0 | K=2) for
// a 16-column chunk whose d-mapping is d(n) = 64j + 4n + c.  A second load at
// row+1 gives VGPR1 (K=1 | K=3).  The A-fragment is the 4 softmax weights
// broadcast (VGPR0 = p0|p2, VGPR1 = p1|p3).  D (replicated over M) VGPR0 is
// the ctx partial for those 16 d's, accumulated in the same permuted layout
// and un-permuted only at the final LDS combine.
//
// LDS map (floats): [0, 32768)  : 8 wave stage buffers (4096 each = 16KB)
//                   [32768,33792): decoder_h copy (1024)
//                   [33792,33808): per-wave {m, Z}
// The 8x1024 ctx combine buffer reuses [0, 8192) after the main loop.
// ---------------------------------------------------------------------------
__global__ __launch_bounds__(256, 1)
__attribute__((amdgpu_num_vgpr(256)))
void attn_partial(
    const float* __restrict__ enc, const float* __restrict__ dec,
    float* __restrict__ rawScores,        // d_out weights region: raw scores
    float* __restrict__ ctxDst,           // wsCtx (split>1) or outCtx (split==1)
    float* __restrict__ wsMZ,             // per-part {m, Z}; unused if split==1
    float* __restrict__ outW_final,       // for split==1 in-kernel finalize
    int Lpart, int groupsPerWave)
{
  extern __shared__ float smem[];
  const int split = L_N / Lpart;
  const int b    = blockIdx.x / split;
  const int part = blockIdx.x % split;
  const int tid  = threadIdx.x;
  const int wave = tid >> 5;
  const int lane = tid & 31;
  const int hi16 = lane >> 4;             // 0: lanes 0-15, 1: lanes 16-31
  const int n16  = lane & 15;

  const float* encB = enc + (size_t)b * (L_N * D_N);
  const float* hB   = dec + (size_t)b * D_N;

  float* stage = smem + wave * 4096;      // 4096 floats = 16KB per wave
  float* hlds  = smem + 32768;            // identity copy of decoder_h[b]
  float* mzsh  = smem + 33792;

  // Stage decoder_h into LDS once (256 threads x float4 = 1024 floats).
  *(v4f*)(hlds + tid * 4) = *(const v4f*)(hB + tid * 4);
  __syncthreads();

  float ctxr[64];                         // ctx acc, frag (j,c) <-> d=64j+4n16+c
#pragma unroll
  for (int i = 0; i < 64; ++i) ctxr[i] = 0.0f;
  float mRun = -__builtin_inff();
  float zRun = 0.0f;

  const int rowBase = part * Lpart + wave * (groupsPerWave * 4);

  for (int g = 0; g < groupsPerWave; ++g) {
    const int r0 = rowBase + g * 4;
    // rows r0 / r0+2 in lane halves; pU = rows r0+1 / r0+3
    const float* pV = encB + (size_t)(r0 + hi16 * 2) * D_N + n16 * 4;
    const float* pU = pV + D_N;

    // ---- Phase A: stream 4 rows (16 KB), dot vs h (from LDS), stash tile ----
    float acc0 = 0.0f, acc1 = 0.0f;
#pragma unroll
    for (int j = 0; j < 16; ++j) {
      v4f v = __builtin_nontemporal_load((const v4f*)(pV + j * 64));
      v4f u = __builtin_nontemporal_load((const v4f*)(pU + j * 64));
      v4f h4 = *(const v4f*)(hlds + j * 64 + n16 * 4);   // broadcast read
      acc0 = fmaf(v.x, h4.x, acc0); acc0 = fmaf(v.y, h4.y, acc0);
      acc0 = fmaf(v.z, h4.z, acc0); acc0 = fmaf(v.w, h4.w, acc0);
      acc1 = fmaf(u.x, h4.x, acc1); acc1 = fmaf(u.y, h4.y, acc1);
      acc1 = fmaf(u.z, h4.z, acc1); acc1 = fmaf(u.w, h4.w, acc1);
      *(v4f*)(stage + (j * 2 + 0) * 128 + lane * 4) = v;
      *(v4f*)(stage + (j * 2 + 1) * 128 + lane * 4) = u;
      if ((j & 3) == 3) __builtin_amdgcn_sched_barrier(0);
    }

    // ---- Phase B: 16-lane reductions -> 4 scores, online softmax ----
#pragma unroll
    for (int sft = 1; sft < 16; sft <<= 1) {
      acc0 += __shfl_xor(acc0, sft, 32);
      acc1 += __shfl_xor(acc1, sft, 32);
    }
    float sc0 = __shfl(acc0, 0, 32);   // row r0
    float sc2 = __shfl(acc0, 16, 32);  // row r0+2
    float sc1 = __shfl(acc1, 0, 32);   // row r0+1
    float sc3 = __shfl(acc1, 16, 32);  // row r0+3
    if (lane < 4) {
      float sv = (lane == 0) ? sc0 : (lane == 1) ? sc1 : (lane == 2) ? sc2 : sc3;
      rawScores[(size_t)b * L_N + r0 + lane] = sv;
    }
    float mNew = fmaxf(fmaxf(fmaxf(sc0, sc1), fmaxf(sc2, sc3)), mRun);
    if (mNew > mRun) {                  // wave-uniform, taken ~log(rows) times
      float scale = __expf(mRun - mNew);
      zRun *= scale;
#pragma unroll
      for (int i = 0; i < 64; ++i) ctxr[i] *= scale;
      mRun = mNew;
    }
    float p0 = __expf(sc0 - mRun), p1 = __expf(sc1 - mRun);
    float p2 = __expf(sc2 - mRun), p3 = __expf(sc3 - mRun);
    zRun += (p0 + p1) + (p2 + p3);

    // ---- Phase C: WMMA accumulate ctx += sum_k p_k * enc[r0+k, :] ----
    v2f afrag;                          // A 16x4: VGPR0 = K0|K2, VGPR1 = K1|K3
    afrag.x = hi16 ? p2 : p0;
    afrag.y = hi16 ? p3 : p1;
    const v8f czero = {};
#pragma unroll
    for (int j = 0; j < 16; ++j) {
      v4f v = *(const v4f*)(stage + (j * 2 + 0) * 128 + lane * 4);
      v4f u = *(const v4f*)(stage + (j * 2 + 1) * 128 + lane * 4);
      v2f bb0; bb0.x = v.x; bb0.y = u.x;   // B 4x16: VGPR0 = K0|K2, VGPR1 = K1|K3
      v2f bb1; bb1.x = v.y; bb1.y = u.y;
      v2f bb2; bb2.x = v.z; bb2.y = u.z;
      v2f bb3; bb3.x = v.w; bb3.y = u.w;
      v8f dA;
      dA = __builtin_amdgcn_wmma_f32_16x16x4_f32(false, afrag, false, bb0, (short)0, czero, false, false);
      ctxr[j * 4 + 0] += dA[0];
      dA = __builtin_amdgcn_wmma_f32_16x16x4_f32(false, afrag, false, bb1, (short)0, czero, false, false);
      ctxr[j * 4 + 1] += dA[0];
      dA = __builtin_amdgcn_wmma_f32_16x16x4_f32(false, afrag, false, bb2, (short)0, czero, false, false);
      ctxr[j * 4 + 2] += dA[0];
      dA = __builtin_amdgcn_wmma_f32_16x16x4_f32(false, afrag, false, bb3, (short)0, czero, false, false);
      ctxr[j * 4 + 3] += dA[0];
      if ((j & 3) == 3) __builtin_amdgcn_sched_barrier(0);
    }
  }

  // ---- Deterministic cross-wave combine (fixed-order sums, no atomics) ----
  if (lane == 0) { mzsh[wave * 2] = mRun; mzsh[wave * 2 + 1] = zRun; }
  __syncthreads();
  float M = mzsh[0];
#pragma unroll
  for (int w = 1; w < WAVES; ++w) M = fmaxf(M, mzsh[w * 2]);
  float Zs = 0.0f;
#pragma unroll
  for (int w = 0; w < WAVES; ++w) Zs += mzsh[w * 2 + 1] * __expf(mzsh[w * 2] - M);
  float fw = __expf(mRun - M);

  // Un-permute: frag (j,c), lane n16 -> d = 64j + 4*n16 + c.  Stage region is
  // dead now; reuse it as the 8x1024 combine buffer (each wave its own slab).
  if (lane < 16) {
#pragma unroll
    for (int j = 0; j < 16; ++j)
#pragma unroll
      for (int c = 0; c < 4; ++c)
        smem[wave * 1024 + j * 64 + lane * 4 + c] = ctxr[j * 4 + c] * fw;
  }
  __syncthreads();

  const int baseIdx = b * split + part;
  const float outScale = (split == 1) ? (1.0f / Zs) : 1.0f;
  for (int d = tid; d < D_N; d += 256) {
    float s = 0.0f;
#pragma unroll
    for (int w = 0; w < WAVES; ++w) s += smem[w * 1024 + d];
    ctxDst[(size_t)baseIdx * D_N + d] = s * outScale;
  }
  if (split > 1) {
    if (tid == 0) { wsMZ[baseIdx * 2] = M; wsMZ[baseIdx * 2 + 1] = Zs; }
  } else {
    // split==1: this block saw all L rows -> finalize weights here.  Each
    // wave normalizes the rows it wrote (same-wave store->load is ordered).
    float inv = 1.0f / Zs;
    const int nRows = groupsPerWave * 4;
    for (int i = lane; i < nRows; i += 32) {
      size_t idx = (size_t)b * L_N + rowBase + i;
      outW_final[idx] = __expf(outW_final[idx] - M) * inv;
    }
  }
}

// ---------------------------------------------------------------------------
// Kernel 2 (split > 1): flash-combine the SPLIT partials per batch, normalize
// the raw scores in place into softmax weights.
// ---------------------------------------------------------------------------
__global__ __launch_bounds__(256, 1) void attn_finalize(
    const float* __restrict__ wsCtx, const float* __restrict__ wsMZ,
    float* __restrict__ outCtx, float* __restrict__ outW, int split)
{
  const int b = blockIdx.x;
  const int tid = threadIdx.x;
  float M = -__builtin_inff();
#pragma unroll
  for (int p = 0; p < 4; ++p)
    if (p < split) M = fmaxf(M, wsMZ[(b * split + p) * 2]);
  float fac[4];
  float Z = 0.0f;
#pragma unroll
  for (int p = 0; p < 4; ++p) {
    fac[p] = (p < split) ? __expf(wsMZ[(b * split + p) * 2] - M) : 0.0f;
    if (p < split) Z += wsMZ[(b * split + p) * 2 + 1] * fac[p];
  }
  const float inv = 1.0f / Z;
  for (int d = tid; d < D_N; d += 256) {
    float s = 0.0f;
#pragma unroll
    for (int p = 0; p < 4; ++p)
      if (p < split)
        s += wsCtx[((size_t)(b * split + p)) * D_N + d] * fac[p];
    outCtx[(size_t)b * D_N + d] = s * inv;
  }
  for (int l = tid; l < L_N; l += 256) {
    size_t idx = (size_t)b * L_N + l;
    outW[idx] = __expf(outW[idx] - M) * inv;
  }
}

extern "C" void kernel_launch(void* const* d_in, const int* in_sizes, int n_in,
                              void* d_out, int out_size, void* d_ws, size_t ws_size,
                              hipStream_t stream)
{
  (void)in_sizes; (void)n_in; (void)out_size;
  const float* enc = (const float*)d_in[0];   // [B, L, D]
  const float* dec = (const float*)d_in[1];   // [B, D]
  float* out    = (float*)d_out;
  float* outCtx = out;                         // [B, D]
  float* outW   = out + (size_t)B_N * D_N;     // [B, L] (weights, flat)

  // Split L across blocks for parallelism if workspace allows the partials.
  const size_t perSplit = (size_t)B_N * (D_N + 2) * sizeof(float);
  int split = 1;
  if (ws_size >= 4 * perSplit)      split = 4;
  else if (ws_size >= 2 * perSplit) split = 2;

  float* wsCtx = (float*)d_ws;
  float* wsMZ  = wsCtx + (size_t)B_N * split * D_N;
  float* ctxDst = (split > 1) ? wsCtx : outCtx;

  const int Lpart = L_N / split;
  const int groupsPerWave = Lpart / (WAVES * 4);
  const size_t smemBytes = (32768 + 1024 + 16) * sizeof(float);  // 135232 B

  attn_partial<<<dim3(B_N * split), dim3(256), smemBytes, stream>>>(
      enc, dec, outW, ctxDst, wsMZ, outW, Lpart, groupsPerWave);
  if (split > 1) {
    attn_finalize<<<dim3(B_N), dim3(256), 0, stream>>>(
        wsCtx, wsMZ, outCtx, outW, split);
  }
}